// PointNetFeaturePropagation_51780125721456
// MI455X (gfx1250) — compile-verified
//
#include <hip/hip_runtime.h>

// ---------------------------------------------------------------------------
// PointNet Feature Propagation for MI455X (gfx1250, wave32, WMMA f32)
// B=16, N=4096, S=1024, D1=D2=256, MLP=[256,256], IN_CH=512
// ---------------------------------------------------------------------------

typedef __attribute__((ext_vector_type(2))) float v2f;
typedef __attribute__((ext_vector_type(8))) float v8f;

#define BATCH  16
#define NPTS   4096
#define SPTS   1024
#define CIN    512
#define COUT   256
#define MROWS  (BATCH * NPTS)      // 65536
#define BN_EPS 1e-5f

// Pair-row stride for the LDS W tile: 64 cols * 2 floats + 32 pad words.
// 160 mod 64 == 32, so pair p (lanes 0-15) and pair p+1 (lanes 16-31)
// land in disjoint 32-bank halves -> conflict-free ds_load_b64.
#define BS_PSTRIDE 160
#define BS_WORDS   (32 * BS_PSTRIDE)   // one 64-k chunk = 32 k-pairs

// ---------------------------------------------------------------------------
// Kernel 1: 3-nearest-neighbor search + inverse-distance weights.
// ---------------------------------------------------------------------------
__global__ __launch_bounds__(256) void top3_kernel(
    const float* __restrict__ xyz1,   // [B,3,N]
    const float* __restrict__ xyz2,   // [B,3,S]
    int* __restrict__ idx,            // [B*N*3]
    float* __restrict__ wgt)          // [B*N*3]
{
    __shared__ float sx[256], sy[256], sz[256];
    const int b = blockIdx.y;
    const int n = blockIdx.x * 256 + threadIdx.x;

    const float* x1 = xyz1 + (size_t)b * 3 * NPTS;
    const float px = x1[n];
    const float py = x1[NPTS + n];
    const float pz = x1[2 * NPTS + n];

    float d0 = 3.4e38f, d1 = 3.4e38f, d2 = 3.4e38f;
    int   i0 = 0, i1 = 0, i2 = 0;

    const float* x2 = xyz2 + (size_t)b * 3 * SPTS;
    for (int s0 = 0; s0 < SPTS; s0 += 256) {
        __syncthreads();
        sx[threadIdx.x] = x2[s0 + threadIdx.x];
        sy[threadIdx.x] = x2[SPTS + s0 + threadIdx.x];
        sz[threadIdx.x] = x2[2 * SPTS + s0 + threadIdx.x];
        __syncthreads();
        for (int j = 0; j < 256; ++j) {
            const float dx = px - sx[j];
            const float dy = py - sy[j];
            const float dz = pz - sz[j];
            const float d = dx * dx + dy * dy + dz * dz;
            const int   s = s0 + j;
            if (d < d2) {
                if (d < d0)      { d2 = d1; i2 = i1; d1 = d0; i1 = i0; d0 = d; i0 = s; }
                else if (d < d1) { d2 = d1; i2 = i1; d1 = d;  i1 = s; }
                else             { d2 = d;  i2 = s; }
            }
        }
    }

    const float r0 = 1.0f / (d0 + 1e-8f);
    const float r1 = 1.0f / (d1 + 1e-8f);
    const float r2 = 1.0f / (d2 + 1e-8f);
    const float rs = 1.0f / (r0 + r1 + r2);

    const size_t o = ((size_t)b * NPTS + n) * 3;
    idx[o + 0] = i0; idx[o + 1] = i1; idx[o + 2] = i2;
    wgt[o + 0] = r0 * rs; wgt[o + 1] = r1 * rs; wgt[o + 2] = r2 * rs;
}

// ---------------------------------------------------------------------------
// Kernel 2: tiled [B,C,N] -> [B,N,*] transpose.  block = (32,8)
// ---------------------------------------------------------------------------
__global__ void transpose_cn_kernel(
    const float* __restrict__ in,   // [B,C,N]
    float* __restrict__ out,        // [B,N,ostride], written at col ooff+c
    int C, int N, int ostride, int ooff)
{
    __shared__ float tile[32][33];
    const int b  = blockIdx.z;
    const int n0 = blockIdx.x * 32;
    const int c0 = blockIdx.y * 32;
    const int tx = threadIdx.x;
    const int ty = threadIdx.y;

    const float* src = in + (size_t)b * C * N;
    #pragma unroll
    for (int i = 0; i < 32; i += 8)
        tile[ty + i][tx] = src[(size_t)(c0 + ty + i) * N + n0 + tx];  // coalesced in n
    __syncthreads();
    float* dst = out + (size_t)b * N * ostride;
    #pragma unroll
    for (int i = 0; i < 32; i += 8)
        dst[(size_t)(n0 + ty + i) * ostride + ooff + c0 + tx] = tile[tx][ty + i]; // coalesced in c
}

// ---------------------------------------------------------------------------
// Kernel 3: 3-point weighted interpolation into X[:, 256:512].
// ---------------------------------------------------------------------------
__global__ __launch_bounds__(256) void interp_kernel(
    const float* __restrict__ f2t,  // [B,S,256]
    const int* __restrict__ idx,    // [B*N*3]
    const float* __restrict__ wgt,  // [B*N*3]
    float* __restrict__ X)          // [M,512]
{
    const size_t t  = (size_t)blockIdx.x * 256 + threadIdx.x;  // M * 64 threads
    const size_t m  = t >> 6;
    const int    cg = (int)(t & 63) << 2;
    const size_t b  = m >> 12;   // /4096
    const size_t o  = m * 3;

    const int   j0 = idx[o + 0], j1 = idx[o + 1], j2 = idx[o + 2];
    const float w0 = wgt[o + 0], w1 = wgt[o + 1], w2 = wgt[o + 2];

    const float* base = f2t + b * (size_t)SPTS * 256;
    const float4 f0 = *(const float4*)(base + (size_t)j0 * 256 + cg);
    const float4 f1 = *(const float4*)(base + (size_t)j1 * 256 + cg);
    const float4 f2 = *(const float4*)(base + (size_t)j2 * 256 + cg);

    float4 r;
    r.x = w0 * f0.x + w1 * f1.x + w2 * f2.x;
    r.y = w0 * f0.y + w1 * f1.y + w2 * f2.y;
    r.z = w0 * f0.z + w1 * f1.z + w2 * f2.z;
    r.w = w0 * f0.w + w1 * f1.w + w2 * f2.w;
    *(float4*)(X + m * CIN + 256 + cg) = r;
}

// ---------------------------------------------------------------------------
// Kernel 4: f32 WMMA GEMM.  Y[M,Nout] = A[M,K] * W[Nout,K]^T
//
// Block: 256 threads (8 waves) -> 128 rows x 64 cols of C.
// Pipeline:
//   - W tile for the NEXT 64-k chunk is global-loaded into VGPRs during the
//     current chunk's WMMAs (register-buffered), stored into the alternate
//     LDS buffer; single barrier per chunk (double-buffered LDS).
//   - LDS layout is pair-transposed: Bs[p*160 + c*2 + {0,1}] = W[c][2p+{0,1}]
//     so a B fragment is ONE aligned ds_load_b64, bank-conflict-free.
//   - A fragment and the 4 B fragments for step kk+4 are prefetched while
//     step kk's 4 v_wmma_f32_16x16x4_f32 execute.
// ---------------------------------------------------------------------------
__global__ __launch_bounds__(256) void gemm_wmma_kernel(
    const float* __restrict__ A,   // [M,K] row-major
    const float* __restrict__ W,   // [Nout,K] row-major
    float* __restrict__ Y,         // [M,Nout]
    int K, int Nout)
{
    __shared__ float Bs[2][BS_WORDS];   // 2 x 20 KB

    const int tid  = threadIdx.x;
    const int wave = tid >> 5;          // 0..7
    const int lane = tid & 31;
    const int lm   = lane & 15;         // row/col within fragment
    const int lk   = (lane >> 4) << 1;  // 0 or 2 (K sub-offset)
    const int lp   = lk >> 1;           // 0 or 1 (K pair sub-offset)
    const int row0 = blockIdx.x * 128 + wave * 16;
    const int col0 = blockIdx.y * 64;

    // Staging coordinates: each thread owns one W row-slice of 16 k values.
    const int sc   = tid >> 2;          // 0..63  (column within tile)
    const int skk0 = (tid & 3) * 16;    // 0,16,32,48 (k offset within chunk)
    const int sp0  = skk0 >> 1;         // starting k-pair

    v8f acc[4];
    #pragma unroll
    for (int t = 0; t < 4; ++t)
        #pragma unroll
        for (int r = 0; r < 8; ++r) acc[t][r] = 0.0f;

    // Prefetch chunk 0 of W into registers.
    float4 wreg[4];
    {
        const float* src = W + (size_t)(col0 + sc) * K + skk0;
        #pragma unroll
        for (int j = 0; j < 4; ++j)
            wreg[j] = *(const float4*)(src + 4 * j);
    }

    const float* Abase = A + (size_t)(row0 + lm) * K + lk;
    const int bcol2 = (tid & 31) >= 16 ? 0 : 0;  // (kept simple; see lm/lp)
    (void)bcol2;

    int buf = 0;
    for (int k0 = 0; k0 < K; k0 += 64) {
        // ---- store staged W registers into LDS buffer `buf` ----
        {
            float* bs = Bs[buf];
            #pragma unroll
            for (int j = 0; j < 4; ++j) {
                const int p = sp0 + 2 * j;
                bs[(p    ) * BS_PSTRIDE + sc * 2 + 0] = wreg[j].x;
                bs[(p    ) * BS_PSTRIDE + sc * 2 + 1] = wreg[j].y;
                bs[(p + 1) * BS_PSTRIDE + sc * 2 + 0] = wreg[j].z;
                bs[(p + 1) * BS_PSTRIDE + sc * 2 + 1] = wreg[j].w;
            }
        }
        __syncthreads();

        // ---- prefetch NEXT chunk of W into registers (latency hidden) ----
        if (k0 + 64 < K) {
            const float* src = W + (size_t)(col0 + sc) * K + k0 + 64 + skk0;
            #pragma unroll
            for (int j = 0; j < 4; ++j)
                wreg[j] = *(const float4*)(src + 4 * j);
            // warm L2 for the next A chunk as well (global_prefetch_b8)
            __builtin_prefetch(Abase + k0 + 64, 0, 0);
        }

        // ---- compute this chunk ----
        const float* Arow = Abase + k0;
        const float* bs = Bs[buf];

        float2 av = *(const float2*)(Arow);
        v2f bcur[4];
        #pragma unroll
        for (int t = 0; t < 4; ++t)
            bcur[t] = *(const v2f*)&bs[lp * BS_PSTRIDE + (t * 16 + lm) * 2];

        #pragma unroll
        for (int kk = 0; kk < 64; kk += 4) {
            float2 avn;
            v2f bnext[4];
            if (kk < 60) {
                avn = *(const float2*)(Arow + kk + 4);
                const int pn = ((kk + 4) >> 1) + lp;
                #pragma unroll
                for (int t = 0; t < 4; ++t)
                    bnext[t] = *(const v2f*)&bs[pn * BS_PSTRIDE + (t * 16 + lm) * 2];
            }
            v2f a; a[0] = av.x; a[1] = av.y;
            #pragma unroll
            for (int t = 0; t < 4; ++t)
                acc[t] = __builtin_amdgcn_wmma_f32_16x16x4_f32(
                    /*neg_a=*/false, a, /*neg_b=*/false, bcur[t],
                    /*c_mod=*/(short)0, acc[t],
                    /*reuse_a=*/false, /*reuse_b=*/false);
            if (kk < 60) {
                av = avn;
                #pragma unroll
                for (int t = 0; t < 4; ++t) bcur[t] = bnext[t];
            }
        }
        buf ^= 1;
    }

    // C/D layout: VGPR r, lanes 0-15 -> row row0+r; lanes 16-31 -> row row0+8+r
    const int rbase = row0 + ((lane >> 4) << 3);
    #pragma unroll
    for (int t = 0; t < 4; ++t) {
        const int col = col0 + t * 16 + lm;
        #pragma unroll
        for (int r = 0; r < 8; ++r)
            Y[(size_t)(rbase + r) * Nout + col] = acc[t][r];
    }
}

// ---------------------------------------------------------------------------
// Kernel 5: deterministic BN statistics, pass 1 (block = row slice, thread = ch)
// ---------------------------------------------------------------------------
__global__ __launch_bounds__(256) void bn_partial_kernel(
    const float* __restrict__ Y,    // [M,256]
    float* __restrict__ psum,       // [slices,256]
    float* __restrict__ psq,        // [slices,256]
    int rowsPerSlice)
{
    const int c  = threadIdx.x;
    const size_t r0 = (size_t)blockIdx.x * rowsPerSlice;
    float s = 0.0f, q = 0.0f;
    for (int i = 0; i < rowsPerSlice; ++i) {
        const float v = Y[(r0 + i) * COUT + c];
        s += v; q += v * v;
    }
    psum[(size_t)blockIdx.x * COUT + c] = s;
    psq [(size_t)blockIdx.x * COUT + c] = q;
}

// Pass 2: fold partials into per-channel mean and inverse stddev.
__global__ __launch_bounds__(256) void bn_finalize_kernel(
    const float* __restrict__ psum, const float* __restrict__ psq,
    float* __restrict__ mean, float* __restrict__ istd,
    int slices, float invM)
{
    const int c = threadIdx.x;
    float s = 0.0f, q = 0.0f;
    for (int i = 0; i < slices; ++i) {
        s += psum[(size_t)i * COUT + c];
        q += psq [(size_t)i * COUT + c];
    }
    const float m   = s * invM;
    const float var = q * invM - m * m;
    mean[c] = m;
    istd[c] = rsqrtf(var + BN_EPS);
}

// Kernel 6: BN apply + LeakyReLU(0.2) in place (float4 over [M,256]).
__global__ __launch_bounds__(256) void bn_act_kernel(
    float* __restrict__ Y,
    const float* __restrict__ mean, const float* __restrict__ istd,
    const float* __restrict__ gamma, const float* __restrict__ beta)
{
    const size_t i  = (size_t)blockIdx.x * 256 + threadIdx.x;  // M*64 vec4
    const int    cg = (int)(i & 63) * 4;
    float4 v = ((const float4*)Y)[i];
    float4 r;
    r.x = (v.x - mean[cg + 0]) * istd[cg + 0] * gamma[cg + 0] + beta[cg + 0];
    r.y = (v.y - mean[cg + 1]) * istd[cg + 1] * gamma[cg + 1] + beta[cg + 1];
    r.z = (v.z - mean[cg + 2]) * istd[cg + 2] * gamma[cg + 2] + beta[cg + 2];
    r.w = (v.w - mean[cg + 3]) * istd[cg + 3] * gamma[cg + 3] + beta[cg + 3];
    r.x = r.x >= 0.0f ? r.x : 0.2f * r.x;
    r.y = r.y >= 0.0f ? r.y : 0.2f * r.y;
    r.z = r.z >= 0.0f ? r.z : 0.2f * r.z;
    r.w = r.w >= 0.0f ? r.w : 0.2f * r.w;
    ((float4*)Y)[i] = r;
}

// ---------------------------------------------------------------------------
// Kernel 7: final BN + LeakyReLU fused into LDS-tiled transpose
// Y[M,256] (row = b*N+n) -> out[B,256,N].  block = (32,8)
// ---------------------------------------------------------------------------
__global__ void bn_act_transpose_out_kernel(
    const float* __restrict__ Y, float* __restrict__ out,
    const float* __restrict__ mean, const float* __restrict__ istd,
    const float* __restrict__ gamma, const float* __restrict__ beta)
{
    __shared__ float tile[32][33];   // [n_local][c_local]
    const int b  = blockIdx.z;
    const int n0 = blockIdx.x * 32;
    const int c0 = blockIdx.y * 32;
    const int tx = threadIdx.x;
    const int ty = threadIdx.y;

    const float* src = Y + (size_t)b * NPTS * COUT;
    #pragma unroll
    for (int i = 0; i < 32; i += 8) {
        const int c = c0 + tx;
        const int n = n0 + ty + i;
        float v = src[(size_t)n * COUT + c];                  // coalesced in c
        v = (v - mean[c]) * istd[c] * gamma[c] + beta[c];
        tile[ty + i][tx] = v >= 0.0f ? v : 0.2f * v;
    }
    __syncthreads();
    float* dst = out + (size_t)b * COUT * NPTS;
    #pragma unroll
    for (int i = 0; i < 32; i += 8)
        dst[(size_t)(c0 + ty + i) * NPTS + n0 + tx] = tile[tx][ty + i]; // coalesced in n
}

// ---------------------------------------------------------------------------
// Host-side launcher
// ---------------------------------------------------------------------------
extern "C" void kernel_launch(void* const* d_in, const int* in_sizes, int n_in,
                              void* d_out, int out_size, void* d_ws, size_t ws_size,
                              hipStream_t stream)
{
    const float* xyz1    = (const float*)d_in[0];
    const float* xyz2    = (const float*)d_in[1];
    const float* points1 = (const float*)d_in[2];
    const float* points2 = (const float*)d_in[3];
    const float* w0      = (const float*)d_in[4];
    // d_in[5] = b0: conv bias cancels exactly under train-mode batch-norm.
    const float* gamma0  = (const float*)d_in[6];
    const float* beta0   = (const float*)d_in[7];
    const float* w1      = (const float*)d_in[8];
    // d_in[9] = b1: cancels likewise.
    const float* gamma1  = (const float*)d_in[10];
    const float* beta1   = (const float*)d_in[11];
    float* out = (float*)d_out;

    char* ws = (char*)d_ws;
    int*   idx  = (int*)  (ws + (0ull));               //  768 KB
    float* wgt  = (float*)(ws + (1ull << 20));         //  768 KB
    float* psum = (float*)(ws + (2ull << 20));         //  256 KB
    float* psq  = (float*)(ws + (2ull << 20) + (1u << 18));
    float* mean = (float*)(ws + (2ull << 20) + (2u << 18));
    float* istd = (float*)(ws + (2ull << 20) + (2u << 18) + 4096);
    float* f2t  = (float*)(ws + (4ull << 20));         //  16 MB  [B,S,256]
    float* X    = (float*)(ws + (32ull << 20));        // 128 MB  [M,512]
    float* Y1   = (float*)(ws + (168ull << 20));       //  64 MB  [M,256]
    float* Y2   = X;                                   // reuse X region after GEMM1

    const dim3 tb32x8(32, 8);

    // 1) 3-NN + weights
    top3_kernel<<<dim3(NPTS / 256, BATCH), 256, 0, stream>>>(xyz1, xyz2, idx, wgt);

    // 2) points1^T -> X[:, 0:256]; points2^T -> f2t
    transpose_cn_kernel<<<dim3(NPTS / 32, 256 / 32, BATCH), tb32x8, 0, stream>>>(
        points1, X, 256, NPTS, CIN, 0);
    transpose_cn_kernel<<<dim3(SPTS / 32, 256 / 32, BATCH), tb32x8, 0, stream>>>(
        points2, f2t, 256, SPTS, 256, 0);

    // 3) interpolation -> X[:, 256:512]
    interp_kernel<<<(MROWS * 64) / 256, 256, 0, stream>>>(f2t, idx, wgt, X);

    // 4) layer 1: GEMM (WMMA f32) + BN + LeakyReLU
    gemm_wmma_kernel<<<dim3(MROWS / 128, COUT / 64), 256, 0, stream>>>(X, w0, Y1, CIN, COUT);
    bn_partial_kernel<<<256, 256, 0, stream>>>(Y1, psum, psq, MROWS / 256);
    bn_finalize_kernel<<<1, 256, 0, stream>>>(psum, psq, mean, istd, 256, 1.0f / (float)MROWS);
    bn_act_kernel<<<(MROWS * 64) / 256, 256, 0, stream>>>(Y1, mean, istd, gamma0, beta0);

    // 5) layer 2: GEMM (WMMA f32) + BN + LeakyReLU fused with final transpose
    gemm_wmma_kernel<<<dim3(MROWS / 128, COUT / 64), 256, 0, stream>>>(Y1, w1, Y2, COUT, COUT);
    bn_partial_kernel<<<256, 256, 0, stream>>>(Y2, psum, psq, MROWS / 256);
    bn_finalize_kernel<<<1, 256, 0, stream>>>(psum, psq, mean, istd, 256, 1.0f / (float)MROWS);
    bn_act_transpose_out_kernel<<<dim3(NPTS / 32, 256 / 32, BATCH), tb32x8, 0, stream>>>(
        Y2, out, mean, istd, gamma1, beta1);
}